// unit_tan_6811818131978
// MI455X (gfx1250) — compile-verified
//
#include <hip/hip_runtime.h>

typedef __attribute__((ext_vector_type(2))) float v2f;
typedef __attribute__((ext_vector_type(8))) float v8f;

namespace {
constexpr int CIN  = 64;
constexpr int COUT = 128;
constexpr int TT   = 256;
constexpr int VV   = 25;
constexpr int KW   = 9;
constexpr int PAD  = 4;
constexpr int TB   = 32;              // output frames per workgroup
constexpr int FRM  = TB + 2 * PAD;    // 40 frames incl halo
constexpr int COLS = FRM * VV;        // 1000 real gemm columns (64 tiles padded)
constexpr int FS   = 26;              // LDS frame stride (dwords, even -> b64 ok)
constexpr int RS   = 1044;            // LDS channel stride (>=40*26, 8*RS%64==32)
constexpr int NCH  = 64;              // output channels per workgroup
constexpr int CHSTR = TT * VV;        // x channel stride in elements (6400)
constexpr int STGW  = TB * VV;        // 800 dwords staging per wave
}

__global__ __launch_bounds__(256)
void stgcn_attn_fused(const float* __restrict__ x,
                      const float* __restrict__ w,
                      const float* __restrict__ bconv,
                      const float* __restrict__ gamma,
                      const float* __restrict__ beta,
                      const float* __restrict__ rmean,
                      const float* __restrict__ rvar,
                      const float* __restrict__ att0,
                      float* __restrict__ out)
{
  extern __shared__ float smem[];
  float*  ylds = smem;                          // NCH * RS floats
  float2* scsh = (float2*)(smem + NCH * RS);    // NCH float2 (scale, shift)
  float*  stgb = smem + NCH * RS + 2 * NCH;     // 8 * STGW store-staging

  const int tid   = threadIdx.x;
  const int lane  = tid & 31;
  const int wave  = tid >> 5;
  const int hl    = lane & 15;   // lane within half-wave
  const int khalf = lane >> 4;   // which K pair this lane holds

  const int tb = blockIdx.x;     // T block (0..7)
  const int cb = blockIdx.y;     // channel half (0..1)
  const int n  = blockIdx.z;     // batch (0..15)
  const int tstart = tb * TB;

  // Fold conv bias into BN: y = dot*scale + ((b - mean)*scale + beta)
  if (tid < NCH) {
    const int o = cb * NCH + tid;
    const float sc = gamma[o] * rsqrtf(rvar[o] + 1e-5f);
    const float sh = (bconv[o] - rmean[o]) * sc + beta[o];
    scsh[tid] = make_float2(sc, sh);
  }
  __syncthreads();

  //============ Stage 1: Y = BN(W @ X + b) via V_WMMA_F32_16X16X4_F32 -> LDS
  // A 16x4 f32 layout: lanes 0-15 rows M0-15 (K pair 0/1), lanes 16-31 (K 2/3)
  v2f Af[4][16];
#pragma unroll
  for (int mt = 0; mt < 4; ++mt) {
    const int o = cb * NCH + mt * 16 + hl;
#pragma unroll
    for (int ks = 0; ks < 16; ++ks) {
      const float2 aw = *(const float2*)(w + o * CIN + ks * 4 + khalf * 2);
      v2f a; a.x = aw.x; a.y = aw.y;
      Af[mt][ks] = a;
    }
  }

  // BN coefficients for the 32 (mt,j) D elements this lane owns (tile-invariant)
  float2 ssr[4][8];
#pragma unroll
  for (int mt = 0; mt < 4; ++mt)
#pragma unroll
    for (int j = 0; j < 8; ++j)
      ssr[mt][j] = scsh[mt * 16 + j + 8 * khalf];

  const float* xn = x + (size_t)n * CIN * CHSTR;

#pragma unroll 2
  for (int it = 0; it < 8; ++it) {            // 64 padded tiles, 8 per wave
    const int tile = wave + it * 8;
    const int col  = tile * 16 + hl;          // gemm column = (frame, v)
    const int f    = col / VV;
    const int v    = col - f * VV;
    const int t    = tstart - PAD + f;        // real time incl halo
    const bool tok = (t >= 0) && (t < TT);
    const int  tc  = t < 0 ? 0 : (t >= TT ? TT - 1 : t);  // safe clamp
    const float* xb = xn + tc * VV + v;

    // Unconditional loads from clamped addresses: invalid columns only
    // pollute their own D column, which is never stored.
    v2f Bf[16];
#pragma unroll
    for (int ks = 0; ks < 16; ++ks) {
      const int c0 = ks * 4 + khalf * 2;
      v2f b;
      b.x = xb[(c0    ) * CHSTR];
      b.y = xb[(c0 + 1) * CHSTR];
      Bf[ks] = b;
    }

    const v8f z = {0.f, 0.f, 0.f, 0.f, 0.f, 0.f, 0.f, 0.f};
    v8f acc[4];
#pragma unroll
    for (int mt = 0; mt < 4; ++mt) acc[mt] = z;

#pragma unroll
    for (int ks = 0; ks < 16; ++ks) {
#pragma unroll
      for (int mt = 0; mt < 4; ++mt) {
        acc[mt] = __builtin_amdgcn_wmma_f32_16x16x4_f32(
            false, Af[mt][ks], false, Bf[ks], (short)0, acc[mt], false, false);
      }
    }

    // D 16x16 f32 layout: VGPR j -> row (j + 8*khalf), lane hl -> col
    const int colLds = f * FS + v;
    const float msk = tok ? 1.f : 0.f;        // reference zero-pads post-BN
    if (col < COLS) {
#pragma unroll
      for (int mt = 0; mt < 4; ++mt) {
#pragma unroll
        for (int j = 0; j < 8; ++j) {
          const int ch = mt * 16 + j + 8 * khalf;
          const float yv = (acc[mt][j] * ssr[mt][j].x + ssr[mt][j].y) * msk;
          ylds[ch * RS + colLds] = yv;
        }
      }
    }
  }
  __syncthreads();

  //============ Stage 2: temporal window attention, online softmax
  const float invV = 1.0f / (float)VV;
  float* outn = out + ((size_t)n * COUT + cb * NCH) * TT * VV;
  float* stg  = stgb + wave * STGW;   // wave-private staging (no barrier needed)

  for (int iter = 0; iter < 8; ++iter) {
    const int cloc = wave + iter * 8;   // local channel 0..63 (wave-uniform)
    const int tl   = lane;              // output frame within block, 0..31
    const float* yc = ylds + cloc * RS;

    float center[VV];
    {
      const float* yf = yc + (tl + PAD) * FS;   // 8B-aligned (even stride)
#pragma unroll
      for (int p = 0; p < 12; ++p) {
        const float2 q = *(const float2*)(yf + 2 * p);
        center[2 * p] = q.x; center[2 * p + 1] = q.y;
      }
      center[24] = yf[24];
    }

    float m = -3.0e38f, d = 0.f;
    float accP[VV], accA[VV];
#pragma unroll
    for (int v = 0; v < VV; ++v) { accP[v] = 0.f; accA[v] = 0.f; }

    const float* a0p = att0 + (cb * NCH + cloc) * KW;
#pragma unroll
    for (int u = 0; u < KW; ++u) {
      float fr[VV];
      const float* yf = yc + (tl + u) * FS;
#pragma unroll
      for (int p = 0; p < 12; ++p) {
        const float2 q = *(const float2*)(yf + 2 * p);
        fr[2 * p] = q.x; fr[2 * p + 1] = q.y;
      }
      fr[24] = yf[24];

      float s = 0.f;
#pragma unroll
      for (int v = 0; v < VV; ++v) s = fmaf(center[v], fr[v], s);
      s *= invV;

      const float a0 = a0p[u];
      const float nm = fmaxf(m, s);
      const float r  = __expf(m - nm);
      const float e  = __expf(s - nm);
      d = d * r + e;
      m = nm;
#pragma unroll
      for (int v = 0; v < VV; ++v) {
        accP[v] = fmaf(e, fr[v], accP[v] * r);   // rescaled exp-weighted sum
        accA[v] = fmaf(a0, fr[v], accA[v]);      // att0 contribution
      }
    }

    // Stage the 25-float result in wave-private LDS, then write out with
    // fully coalesced b128 stores (wave block = 800 dwords = 3200B aligned).
    const float invd = 1.0f / d;
#pragma unroll
    for (int v = 0; v < VV; ++v)
      stg[tl * VV + v] = fmaf(accP[v], invd, accA[v]);

    float* opw = outn + ((size_t)cloc * TT + tstart) * VV;  // 16B aligned
#pragma unroll
    for (int c2 = 0; c2 < 6; ++c2) {
      const float4 q = *(const float4*)(stg + c2 * 128 + lane * 4);
      *(float4*)(opw + c2 * 128 + lane * 4) = q;
    }
    opw[768 + lane] = stg[768 + lane];
  }
}

extern "C" void kernel_launch(void* const* d_in, const int* in_sizes, int n_in,
                              void* d_out, int out_size, void* d_ws, size_t ws_size,
                              hipStream_t stream) {
  const float* x   = (const float*)d_in[0];
  const float* w   = (const float*)d_in[1];
  const float* bcv = (const float*)d_in[2];
  const float* gam = (const float*)d_in[3];
  const float* bet = (const float*)d_in[4];
  const float* rm  = (const float*)d_in[5];
  const float* rv  = (const float*)d_in[6];
  const float* a0  = (const float*)d_in[7];
  // d_in[8] (A) and d_in[9] (keep_prob) are unused by the reference.

  const size_t smem =
      (size_t)(NCH * RS + 2 * NCH + 8 * STGW) * sizeof(float); // ~287 KB
  hipFuncSetAttribute((const void*)stgcn_attn_fused,
                      hipFuncAttributeMaxDynamicSharedMemorySize, (int)smem);

  dim3 grid(TT / TB, COUT / NCH, 16);   // (8, 2, 16)
  stgcn_attn_fused<<<grid, 256, smem, stream>>>(
      x, w, bcv, gam, bet, rm, rv, a0, (float*)d_out);
}